// CompositeLoss_63814624084052
// MI455X (gfx1250) — compile-verified
//
#include <hip/hip_runtime.h>
#include <math.h>

// Problem geometry (fixed by the reference: B=8, C=1, H=W=1024, fp32)
#define EPSF   1e-6f
#define BATCH  8
#define HDIM   1024
#define WDIM   1024
#define HW     (HDIM * WDIM)          // 1,048,576
#define NTOT   (BATCH * HW)           // 8,388,608
#define CHUNKS 128                    // blocks per image for chunked reductions
#define PC     (BATCH * CHUNKS)       // 1024 partial slots
#define SBLK   ((WDIM/32) * (HDIM/8)) // 4096 stencil blocks per image
#define PBIG   (BATCH * SBLK)

typedef __attribute__((ext_vector_type(2))) float v2f;
typedef __attribute__((ext_vector_type(8))) float v8f;
typedef __attribute__((ext_vector_type(4))) unsigned int u32x4;
typedef __attribute__((ext_vector_type(4))) int i32x4;
typedef __attribute__((ext_vector_type(8))) int i32x8;

#ifndef __has_builtin
#define __has_builtin(x) 0
#endif
#if __has_builtin(__builtin_amdgcn_tensor_load_to_lds) && \
    __has_builtin(__builtin_amdgcn_s_wait_tensorcnt)
#define HAVE_TDM 1
#else
#define HAVE_TDM 0
#endif

// ---------------------------------------------------------------------------
// Wave-wide sum via V_WMMA_F32_16X16X4_F32 with an all-ones B matrix.
// With a = {v, 0} per lane and B = ones: D[m][n] = v_m + v_{m+16}; summing the
// 8 D vgprs per lane and adding lane0+lane16 yields the full 32-lane sum.
// ---------------------------------------------------------------------------
__device__ __forceinline__ float wave_sum_wmma(float v) {
  v2f a; a[0] = v;    a[1] = 0.0f;
  v2f b; b[0] = 1.0f; b[1] = 1.0f;
  v8f c = {0.f, 0.f, 0.f, 0.f, 0.f, 0.f, 0.f, 0.f};
  c = __builtin_amdgcn_wmma_f32_16x16x4_f32(false, a, false, b, (short)0, c,
                                            false, false);
  float s = c[0] + c[1] + c[2] + c[3] + c[4] + c[5] + c[6] + c[7];
  return __shfl(s, 0, 32) + __shfl(s, 16, 32);
}

// Block reduction: WMMA wave sums -> LDS -> thread 0 writes NTERMS partials,
// term-major with given stride. Deterministic (no atomics).
#define BLOCK_REDUCE_STORE(TID, NTERMS, VALS, DST, STRIDE, PIDX) do {        \
    __shared__ float red_[8][NTERMS];                                        \
    int wv_ = (TID) >> 5, ln_ = (TID) & 31;                                  \
    for (int t_ = 0; t_ < (NTERMS); ++t_) {                                  \
      float s_ = wave_sum_wmma((VALS)[t_]);                                  \
      if (ln_ == 0) red_[wv_][t_] = s_;                                      \
    }                                                                        \
    __syncthreads();                                                         \
    if ((TID) == 0) {                                                        \
      for (int t_ = 0; t_ < (NTERMS); ++t_) {                                \
        float a_ = 0.f;                                                      \
        for (int w_ = 0; w_ < 8; ++w_) a_ += red_[w_][t_];                   \
        (DST)[t_ * (STRIDE) + (PIDX)] = a_;                                  \
      }                                                                      \
    }                                                                        \
  } while (0)

// ---------------------------------------------------------------------------
// K0: probs = sigmoid(logits); dice partials (per image) + bce/focal/fbce
// partials (global). grid = (CHUNKS, BATCH), 256 threads.
// ---------------------------------------------------------------------------
__global__ __launch_bounds__(256)
void k_pointwise(const float* __restrict__ logits,
                 const float* __restrict__ targs,
                 const float* __restrict__ fw,
                 float* __restrict__ probs,
                 float* __restrict__ ptS,   // terms 0..2: sum(p*t), sum(p), sum(t)
                 float* __restrict__ gtP) { // terms 0..2: bce, focal, fbce sums
  const int b = blockIdx.y, chunk = blockIdx.x, tid = threadIdx.x;
  const long base = (long)b * HW + (long)chunk * (HW / CHUNKS);
  const float4* L4 = (const float4*)(logits + base);
  const float4* T4 = (const float4*)(targs + base);
  const float4* F4 = (const float4*)(fw + base);
  float4* P4 = (float4*)(probs + base);
  float s_pt = 0.f, s_p = 0.f, s_t = 0.f, s_b = 0.f, s_f = 0.f, s_fb = 0.f;
  const int n4 = (HW / CHUNKS) / 4;  // 2048
  for (int i = tid; i < n4; i += 256) {
    float4 x4 = L4[i], t4 = T4[i], w4 = F4[i], p4;
    float xs[4] = {x4.x, x4.y, x4.z, x4.w};
    float ts[4] = {t4.x, t4.y, t4.z, t4.w};
    float wv[4] = {w4.x, w4.y, w4.z, w4.w};
    float ps[4];
#pragma unroll
    for (int k = 0; k < 4; ++k) {
      float x = xs[k], t = ts[k];
      float p = 1.f / (1.f + __expf(-x));
      ps[k] = p;
      s_p += p; s_t += t; s_pt += p * t;
      float bce = fmaxf(x, 0.f) - x * t + log1pf(__expf(-fabsf(x)));
      s_b += bce;
      float pt = p * t + (1.f - p) * (1.f - t);
      float at = 0.25f * t + 0.75f * (1.f - t);
      float om = 1.f - pt;
      s_f += -at * om * om * __logf(pt + EPSF);
      s_fb += bce * (1.f + wv[k]);
    }
    p4.x = ps[0]; p4.y = ps[1]; p4.z = ps[2]; p4.w = ps[3];
    P4[i] = p4;
  }
  const int pidx = b * CHUNKS + chunk;
  {
    float vals[3] = {s_pt, s_p, s_t};
    BLOCK_REDUCE_STORE(tid, 3, vals, ptS, PC, pidx);
  }
  {
    float vals[3] = {s_b, s_f, s_fb};
    BLOCK_REDUCE_STORE(tid, 3, vals, gtP, PC, pidx);
  }
}

// ---------------------------------------------------------------------------
// Fused skeletonize round: one kernel computes, per 32x8 output tile,
//   e = erode(x)  (3x3 min, +inf SAME pad)   -> written to dst_e (+cap)
//   o = dilate(e) (3x3 max, -inf SAME pad over *valid* e positions)
//   skel update:  first: skel = relu(x-o) ; else skel += relu(d - skel*d)
// x-tile has a 2-wide halo (36x12). Interior tiles are fetched by the Tensor
// Data Mover (tensor_load_to_lds + s_wait_tensorcnt); border tiles use a
// padded manual fill. Erode-tile slots outside the image are forced to -inf
// so the subsequent dilate sees correct SAME padding of the composition.
// ---------------------------------------------------------------------------
__global__ __launch_bounds__(256)
void k_round(const float* __restrict__ src, float* __restrict__ dst_e,
             float* __restrict__ skel, float* __restrict__ cap, int first) {
  __shared__ float xt[12][36];   // x tile, halo 2   (contiguous: TDM target)
  __shared__ float et[10][34];   // erode tile, halo 1
  const int b = blockIdx.z;
  const int ox = blockIdx.x * 32, oy = blockIdx.y * 8;
  const float* s = src + (long)b * HW;
  const int tid = threadIdx.y * 32 + threadIdx.x;

#if HAVE_TDM
  const bool interior = (ox >= 2) && (ox + 33 < WDIM) && (oy >= 2) && (oy + 9 < HDIM);
  if (interior) {
    if (tid < 32) {  // one wave builds the D# and issues the TDM load
      unsigned long long ga =
          (unsigned long long)(const void*)(s + (long)(oy - 2) * WDIM + (ox - 2));
      unsigned lds = (unsigned)(unsigned long long)(void*)&xt[0][0];
      // D# group0: count=1, lds_addr, global_addr[56:0], type=2
      u32x4 g0;
      g0[0] = 1u;
      g0[1] = lds;
      g0[2] = (unsigned)(ga & 0xffffffffull);
      g0[3] = (unsigned)((ga >> 32) & 0x1ffffffull) | (2u << 30);
      // D# group1: data_size=4B (code 2); tensor dims huge (no OOB for
      // interior tiles); tile 36x12; row stride WDIM.
      const unsigned long long td0 = 0x7fffffffull, td1 = 0x7fffffffull;
      const unsigned long long tile0 = 36ull, tile1 = 12ull;
      const unsigned long long str0 = (unsigned long long)WDIM;
      unsigned long long w0 = 0x00020000ull | ((td0 & 0xffffull) << 48);
      unsigned long long w1 = ((td0 >> 16) & 0xffffull) | (td1 << 16) | (tile0 << 48);
      unsigned long long w2 = tile1 | ((str0 & 0xffffffffull) << 32);
      unsigned long long w3 = (str0 >> 32) & 0xffffull;  // dim1 stride unused (2D)
      i32x8 g1;
      g1[0] = (int)(w0); g1[1] = (int)(w0 >> 32);
      g1[2] = (int)(w1); g1[3] = (int)(w1 >> 32);
      g1[4] = (int)(w2); g1[5] = (int)(w2 >> 32);
      g1[6] = (int)(w3); g1[7] = (int)(w3 >> 32);
      i32x4 g2; g2[0] = 0; g2[1] = 0; g2[2] = 0; g2[3] = 0;
      i32x4 g3 = g2;
#if __clang_major__ >= 23
      i32x8 g4; g4[0]=0; g4[1]=0; g4[2]=0; g4[3]=0; g4[4]=0; g4[5]=0; g4[6]=0; g4[7]=0;
      __builtin_amdgcn_tensor_load_to_lds(g0, g1, g2, g3, g4, 0);
#else
      __builtin_amdgcn_tensor_load_to_lds(g0, g1, g2, g3, 0);
#endif
      __builtin_amdgcn_s_wait_tensorcnt(0);
    }
  } else
#endif
  {
    for (int q = tid; q < 12 * 36; q += 256) {
      int r = q / 36, c = q - r * 36;
      int gy = oy + r - 2, gx = ox + c - 2;
      float v = 3.0e38f;  // +inf pad for the min
      if (gy >= 0 && gy < HDIM && gx >= 0 && gx < WDIM) v = s[gy * WDIM + gx];
      xt[r][c] = v;
    }
  }
  __syncthreads();

  // erode tile: position (r,c) <-> global (oy+r-1, ox+c-1) <-> xt[r+1][c+1]
  for (int q = tid; q < 10 * 34; q += 256) {
    int r = q / 34, c = q - r * 34;
    int gy = oy + r - 1, gx = ox + c - 1;
    float m = -3.0e38f;  // outside image: -inf for the following max
    if (gy >= 0 && gy < HDIM && gx >= 0 && gx < WDIM) {
      m = xt[r][c];
      m = fminf(m, xt[r][c + 1]);     m = fminf(m, xt[r][c + 2]);
      m = fminf(m, xt[r + 1][c]);     m = fminf(m, xt[r + 1][c + 1]);
      m = fminf(m, xt[r + 1][c + 2]); m = fminf(m, xt[r + 2][c]);
      m = fminf(m, xt[r + 2][c + 1]); m = fminf(m, xt[r + 2][c + 2]);
    }
    et[r][c] = m;
  }
  __syncthreads();

  const int tx = threadIdx.x, ty = threadIdx.y;
  float o = et[ty][tx];
  o = fmaxf(o, et[ty][tx + 1]);     o = fmaxf(o, et[ty][tx + 2]);
  o = fmaxf(o, et[ty + 1][tx]);     o = fmaxf(o, et[ty + 1][tx + 1]);
  o = fmaxf(o, et[ty + 1][tx + 2]); o = fmaxf(o, et[ty + 2][tx]);
  o = fmaxf(o, et[ty + 2][tx + 1]); o = fmaxf(o, et[ty + 2][tx + 2]);
  const long idx = (long)b * HW + (long)(oy + ty) * WDIM + (ox + tx);
  float ev = et[ty + 1][tx + 1];     // erode(x) at this pixel
  dst_e[idx] = ev;
  if (cap) cap[idx] = ev;
  float d = fmaxf(xt[ty + 2][tx + 2] - o, 0.f);  // relu(x - open(x))
  if (first) {
    skel[idx] = d;
  } else {
    float sk = skel[idx];
    skel[idx] = sk + fmaxf(d - sk * d, 0.f);
  }
}

// ---------------------------------------------------------------------------
// Boundary dice stencil: pb = relu(dilate(p)-erode(p)), tb likewise; reduce
// sum(pb*tb), sum(pb), sum(tb) per image. One NaN-padded tile serves both
// min and max (v_min/max_num ignore NaN).
// ---------------------------------------------------------------------------
__global__ __launch_bounds__(256)
void k_boundary(const float* __restrict__ probs, const float* __restrict__ targs,
                float* __restrict__ ptB) {
  __shared__ float tp[10][34], tt[10][34];
  const int b = blockIdx.z;
  const int ox = blockIdx.x * 32, oy = blockIdx.y * 8;
  const float* pp = probs + (long)b * HW;
  const float* pt = targs + (long)b * HW;
  const int tid = threadIdx.y * 32 + threadIdx.x;
  const float qnan = __int_as_float(0x7fc00000);
  for (int q = tid; q < 340; q += 256) {
    int r = q / 34, c = q - r * 34;
    int gy = oy + r - 1, gx = ox + c - 1;
    float vp = qnan, vt = qnan;
    if (gy >= 0 && gy < HDIM && gx >= 0 && gx < WDIM) {
      vp = pp[gy * WDIM + gx];
      vt = pt[gy * WDIM + gx];
    }
    tp[r][c] = vp; tt[r][c] = vt;
  }
  __syncthreads();
  const int tx = threadIdx.x, ty = threadIdx.y;
  float pmx = tp[ty + 1][tx + 1], pmn = pmx;  // center is always valid
  float tmx = tt[ty + 1][tx + 1], tmn = tmx;
#pragma unroll
  for (int dr = 0; dr < 3; ++dr)
#pragma unroll
    for (int dc = 0; dc < 3; ++dc) {
      float vp = tp[ty + dr][tx + dc], vt = tt[ty + dr][tx + dc];
      pmx = fmaxf(pmx, vp); pmn = fminf(pmn, vp);
      tmx = fmaxf(tmx, vt); tmn = fminf(tmn, vt);
    }
  float pb = fmaxf(pmx - pmn, 0.f);
  float tb = fmaxf(tmx - tmn, 0.f);
  float vals[3] = {pb * tb, pb, tb};
  const int pidx = b * SBLK + blockIdx.y * gridDim.x + blockIdx.x;
  BLOCK_REDUCE_STORE(tid, 3, vals, ptB, PBIG, pidx);
}

// ---------------------------------------------------------------------------
// TV (valid diffs) + reflect-padded Laplacian |lap| sums (global).
// ---------------------------------------------------------------------------
__global__ __launch_bounds__(256)
void k_tvcurv(const float* __restrict__ probs, float* __restrict__ gtP) {
  const int b = blockIdx.y, chunk = blockIdx.x, tid = threadIdx.x;
  const float* p = probs + (long)b * HW;
  const int cbase = chunk * (HW / CHUNKS);
  float sdx = 0.f, sdy = 0.f, scv = 0.f;
  for (int i = tid; i < HW / CHUNKS; i += 256) {
    int pix = cbase + i;
    int y = pix >> 10, x = pix & 1023;
    float c = p[pix];
    if (y < HDIM - 1) sdx += fabsf(p[pix + WDIM] - c);
    if (x < WDIM - 1) sdy += fabsf(p[pix + 1] - c);
    int ym = (y == 0) ? 1 : (y - 1), yp = (y == HDIM - 1) ? (HDIM - 2) : (y + 1);
    int xm = (x == 0) ? 1 : (x - 1), xp = (x == WDIM - 1) ? (WDIM - 2) : (x + 1);
    float lap = -4.f * c + p[ym * WDIM + x] + p[yp * WDIM + x]
                + p[y * WDIM + xm] + p[y * WDIM + xp];
    scv += fabsf(lap);
  }
  float vals[3] = {sdx, sdy, scv};
  BLOCK_REDUCE_STORE(tid, 3, vals, gtP + 3 * PC, PC, b * CHUNKS + chunk);
}

// ---------------------------------------------------------------------------
// Fused skeleton reductions (8 per-image sums): clDice / cbDice / recall.
// ---------------------------------------------------------------------------
__global__ __launch_bounds__(256)
void k_skelred(const float* __restrict__ sp, const float* __restrict__ sgb,
               const float* __restrict__ probs, const float* __restrict__ targs,
               const float* __restrict__ dp, const float* __restrict__ dg,
               float* __restrict__ ptS) {
  const int b = blockIdx.y, chunk = blockIdx.x, tid = threadIdx.x;
  const long base = (long)b * HW + (long)chunk * (HW / CHUNKS);
  float a0 = 0, a1 = 0, a2 = 0, a3 = 0, a4 = 0, a5 = 0, a6 = 0, a7 = 0;
  for (int i = tid; i < HW / CHUNKS; i += 256) {
    long idx = base + i;
    float spv = sp[idx], sgv = sgb[idx], pv = probs[idx], tv = targs[idx];
    float wp = fminf(fmaxf(dp[idx] + EPSF, 0.f), 1.f);
    float wg = fminf(fmaxf(dg[idx] + EPSF, 0.f), 1.f);
    a0 += spv * tv;       a1 += spv;
    a2 += sgv * pv;       a3 += sgv;
    a4 += spv * tv * wg;  a5 += spv * wg;
    a6 += sgv * pv * wp;  a7 += sgv * wp;
  }
  float vals[8] = {a0, a1, a2, a3, a4, a5, a6, a7};
  BLOCK_REDUCE_STORE(tid, 8, vals, ptS + 3 * PC, PC, b * CHUNKS + chunk);
}

// ---------------------------------------------------------------------------
// Final combine: one block; wave b sums image-b partials (WMMA wave sums),
// thread 0 evaluates every loss term and writes the scalar total.
// ---------------------------------------------------------------------------
__global__ __launch_bounds__(256)
void k_combine(const float* __restrict__ ptS, const float* __restrict__ ptB,
               const float* __restrict__ gt, float* __restrict__ out) {
  __shared__ float simg[8][16];
  __shared__ float sglb[8];
  const int wave = threadIdx.x >> 5, lane = threadIdx.x & 31;
  const int b = wave;
  for (int t = 0; t < 11; ++t) {
    const float* arr = ptS + t * PC + b * CHUNKS;
    float l = 0.f;
    for (int i = lane; i < CHUNKS; i += 32) l += arr[i];
    float s = wave_sum_wmma(l);
    if (lane == 0) simg[b][t] = s;
  }
  for (int t = 0; t < 3; ++t) {
    const float* arr = ptB + t * PBIG + b * SBLK;
    float l = 0.f;
    for (int i = lane; i < SBLK; i += 32) l += arr[i];
    float s = wave_sum_wmma(l);
    if (lane == 0) simg[b][11 + t] = s;
  }
  if (wave < 6) {
    const float* arr = gt + wave * PC;
    float l = 0.f;
    for (int i = lane; i < PC; i += 32) l += arr[i];
    float s = wave_sum_wmma(l);
    if (lane == 0) sglb[wave] = s;
  }
  __syncthreads();
  if (threadIdx.x == 0) {
    float l_dice = 0.f, l_cl = 0.f, l_cb = 0.f, l_dbc = 0.f, l_sk = 0.f;
    for (int k = 0; k < 8; ++k) {
      float inter = simg[k][0], sump = simg[k][1], sumt = simg[k][2];
      l_dice += 1.f - (2.f * inter + 1.f) / (sump + sumt + 1.f + EPSF);
      float tprec = (simg[k][3] + EPSF) / (simg[k][4] + EPSF);
      float tsens = (simg[k][5] + EPSF) / (simg[k][6] + EPSF);
      l_cl += 1.f - (2.f * tprec * tsens + 1.f) / (tprec + tsens + 1.f);
      float cbp = (simg[k][7] + EPSF) / (simg[k][8] + EPSF);
      float cbs = (simg[k][9] + EPSF) / (simg[k][10] + EPSF);
      l_cb += 1.f - (2.f * cbp * cbs + 1.f) / (cbp + cbs + 1.f);
      l_dbc += 1.f - (2.f * simg[k][11] + 1.f)
                      / (simg[k][12] + simg[k][13] + 1.f + EPSF);
      l_sk += 1.f - simg[k][5] / (simg[k][6] + EPSF);
    }
    l_dice *= 0.125f; l_cl *= 0.125f; l_cb *= 0.125f;
    l_dbc *= 0.125f;  l_sk *= 0.125f;
    const float Nf = (float)NTOT;
    float l_bce = sglb[0] / Nf;
    float l_foc = sglb[1] / Nf;
    float l_fb  = sglb[2] / Nf;
    float l_tv  = sglb[3] / (8.f * (HDIM - 1) * WDIM)
                + sglb[4] / (8.f * HDIM * (WDIM - 1));
    float l_cv  = sglb[5] / Nf;
    out[0] = l_dice + l_bce + l_foc + l_cl + l_dbc + l_cb + l_sk + l_fb
             + l_tv + l_cv;
  }
}

// ---------------------------------------------------------------------------
// Host orchestration. ws layout (floats):
//   [0,N)   probs      [N,2N)  bufA      [2N,3N) bufB
//   [3N,4N) sp         [4N,5N) sg        [5N,6N) distP    [6N,7N) distG
//   then partial arrays (11*PC + 3*PBIG + 6*PC floats, < 0.5 MB)
// Total ~235 MB.
// ---------------------------------------------------------------------------
extern "C" void kernel_launch(void* const* d_in, const int* in_sizes, int n_in,
                              void* d_out, int out_size, void* d_ws, size_t ws_size,
                              hipStream_t stream) {
  (void)in_sizes; (void)n_in; (void)out_size; (void)ws_size;
  const float* logits = (const float*)d_in[0];
  const float* targs  = (const float*)d_in[1];
  const float* fw     = (const float*)d_in[2];
  float* out = (float*)d_out;
  float* w = (float*)d_ws;

  const long N = (long)NTOT;
  float* probs = w;
  float* bufA  = w + N;
  float* bufB  = w + 2 * N;
  float* spb   = w + 3 * N;
  float* sgb   = w + 4 * N;
  float* dp    = w + 5 * N;
  float* dg    = w + 6 * N;
  float* ptS   = w + 7 * N;            // 11 * PC
  float* ptB   = ptS + 11 * PC;        // 3 * PBIG
  float* gtP   = ptB + 3 * PBIG;       // 6 * PC

  dim3 rb(256), rg(CHUNKS, BATCH);
  dim3 sb(32, 8), sgrid(WDIM / 32, HDIM / 8, BATCH);

  // probs + dice/bce/focal/fbce partials
  k_pointwise<<<rg, rb, 0, stream>>>(logits, targs, fw, probs, ptS, gtP);

  // soft_skeletonize(probs) -> spb ; capture erode^3 -> dp
  {
    const float* x = probs; float* e = bufA;
    for (int r = 0; r < 10; ++r) {
      k_round<<<sgrid, sb, 0, stream>>>(x, e, spb, (r == 2) ? dp : nullptr,
                                        (r == 0) ? 1 : 0);
      x = e; e = (e == bufA) ? bufB : bufA;
    }
  }
  // soft_skeletonize(targets) -> sgb ; capture erode^3 -> dg
  {
    const float* x = targs; float* e = bufA;
    for (int r = 0; r < 10; ++r) {
      k_round<<<sgrid, sb, 0, stream>>>(x, e, sgb, (r == 2) ? dg : nullptr,
                                        (r == 0) ? 1 : 0);
      x = e; e = (e == bufA) ? bufB : bufA;
    }
  }

  // boundary dice, TV/curvature, fused skeleton sums
  k_boundary<<<sgrid, sb, 0, stream>>>(probs, targs, ptB);
  k_tvcurv<<<rg, rb, 0, stream>>>(probs, gtP);
  k_skelred<<<rg, rb, 0, stream>>>(spb, sgb, probs, targs, dp, dg, ptS);

  // deterministic final combine
  k_combine<<<1, 256, 0, stream>>>(ptS, ptB, gtP, out);
}